// WeightedAtomLayer_55379308314866
// MI455X (gfx1250) — compile-verified
//
#include <hip/hip_runtime.h>
#include <math.h>

// WeightedAtomLayer on gfx1250:
//   s_n = sum_k values[idx[n,k], :]          (irregular gather, L2-resident)
//   y_n = tanh(W_n @ s_n + b_n)              (per-neuron 16x16 matvec -> WMMA)
//
// Memory-bound: ~670 MB HBM traffic, dominated by streaming W (512 MB).
// One wave (32 lanes) per neuron; V_WMMA_F32_16X16X4_F32 chained 4x over K.

typedef __attribute__((ext_vector_type(2))) float v2f;
typedef __attribute__((ext_vector_type(4))) float v4f;
typedef __attribute__((ext_vector_type(8))) float v8f;

#define DIM  16
#define FANK 8

__global__ __launch_bounds__(256)
void watom_wmma_kernel(const float* __restrict__ values,
                       const int*   __restrict__ idx,
                       const float* __restrict__ W,
                       const float* __restrict__ bias,
                       float*       __restrict__ out,
                       int N)
{
    const int lane          = threadIdx.x & 31;   // wave32
    const int waveInBlock   = threadIdx.x >> 5;
    const int wavesPerBlock = blockDim.x >> 5;
    const int waveId        = blockIdx.x * wavesPerBlock + waveInBlock;
    const int numWaves      = gridDim.x * wavesPerBlock;

    const int hi  = lane >> 4;    // 0 = lower half-wave, 1 = upper half-wave
    const int col = lane & 15;    // matrix row (A) / column (B,D) owned by lane

    for (int n = waveId; n < N; n += numWaves) {
        // Prefetch next iteration's W_n (1 KB): 32 lanes x 32 B covers it.
        if (n + numWaves < N) {
            const char* pw = (const char*)(W + (size_t)(n + numWaves) * (DIM * DIM));
            __builtin_prefetch(pw + lane * 32, 0, 1);   // -> global_prefetch_b8
        }

        // ---- gather + fan-in sum: s[d] = sum_{k<8} values[idx[n][k]][d] ----
        // lane sums rows k = {hi, 2+hi, 4+hi, 6+hi} at column d = col;
        // each vector load covers two full 64 B value-rows (coalesced).
        const int* idxr = idx + n * FANK;
        float acc = 0.0f;
#pragma unroll
        for (int i = 0; i < 4; ++i) {
            const int  k   = 2 * i + hi;
            const long row = (long)idxr[k];            // broadcast within half
            acc += values[row * DIM + col];
        }
        // fold the two halves: afterwards every lane holds s[col]
        acc += __shfl_xor(acc, 16, 32);

        // ---- per-neuron matvec y = W_n @ s via V_WMMA_F32_16X16X4_F32 ----
        // A: 16x4 f32 slice of W_n. ISA layout: lane(l) VGPR0 = A[l&15][k0+2*hi],
        //    VGPR1 = A[l&15][k0+2*hi+1] -> one aligned v2f load from row-major W.
        // B: 4x16 with every column = s-slice, so every column of D equals y.
        const float* Wn = W + (size_t)n * (DIM * DIM);
        v8f c = {};
#pragma unroll
        for (int k0 = 0; k0 < DIM; k0 += 4) {
            v2f a = *(const v2f*)(Wn + col * DIM + k0 + 2 * hi);
            v2f bf;
            bf.x = __shfl(acc, k0 + 2 * hi,     32);   // B[k0+2hi][*]   = s[k0+2hi]
            bf.y = __shfl(acc, k0 + 2 * hi + 1, 32);   // B[k0+2hi+1][*] = s[k0+2hi+1]
            c = __builtin_amdgcn_wmma_f32_16x16x4_f32(
                    /*neg_a=*/false, a, /*neg_b=*/false, bf,
                    /*c_mod=*/(short)0, c, /*reuse_a=*/false, /*reuse_b=*/false);
        }

        // D layout: lane in half `hi` holds y[8*hi + r] in c[r] (all columns
        // identical). Lanes 0 and 16 each emit 8 outputs (bias + tanh).
        if (col == 0) {
            const float* bp = bias + (size_t)n * DIM + hi * 8;
            float*       op = out  + (size_t)n * DIM + hi * 8;
            float yv[8];
#pragma unroll
            for (int r = 0; r < 8; ++r)
                yv[r] = tanhf(c[r] + bp[r]);
            v4f o0 = {yv[0], yv[1], yv[2], yv[3]};
            v4f o1 = {yv[4], yv[5], yv[6], yv[7]};
            *(v4f*)(op)     = o0;                      // global_store_b128
            *(v4f*)(op + 4) = o1;
        }
    }
}

extern "C" void kernel_launch(void* const* d_in, const int* in_sizes, int n_in,
                              void* d_out, int out_size, void* d_ws, size_t ws_size,
                              hipStream_t stream)
{
    // setup_inputs order: values[M,16] f32, idx[N,8] i32, W[N,16,16] f32, b[N,16] f32
    const float* values = (const float*)d_in[0];
    const int*   idx    = (const int*)  d_in[1];
    const float* W      = (const float*)d_in[2];
    const float* bias   = (const float*)d_in[3];
    float*       out    = (float*)d_out;

    const int N = in_sizes[3] / DIM;   // 500,000

    // 4096 blocks x 8 waves = 32768 waves in flight; grid-stride ~15 neurons
    // per wave keeps the W stream saturated and lets the prefetch run ahead.
    const int threads = 256;
    int blocks = 4096;
    const int wavesTotalNeeded = (N + 0);
    if (blocks * (threads / 32) > wavesTotalNeeded) {
        blocks = (wavesTotalNeeded + (threads / 32) - 1) / (threads / 32);
        if (blocks < 1) blocks = 1;
    }

    watom_wmma_kernel<<<blocks, threads, 0, stream>>>(values, idx, W, bias, out, N);
}